// exp_inference_90340342104681
// MI455X (gfx1250) — compile-verified
//
#include <hip/hip_runtime.h>
#include <hip/hip_bf16.h>
#include <stdint.h>

typedef unsigned int       u32;
typedef unsigned long long u64;
typedef u32   u32x4 __attribute__((ext_vector_type(4)));
typedef int   i32x4 __attribute__((ext_vector_type(4)));
typedef int   i32x8 __attribute__((ext_vector_type(8)));
typedef float f32x4 __attribute__((ext_vector_type(4)));

#define TILE_COLS  256
#define TILE_ROWS  16
#define TILE_ELEMS (TILE_COLS * TILE_ROWS)   // 4096 floats = 16KB
#define TILE_VEC4  (TILE_ELEMS / 4)          // 1024 float4s
#define BLOCK      256

#if __has_builtin(__builtin_amdgcn_tensor_load_to_lds) && __has_builtin(__builtin_amdgcn_s_wait_tensorcnt)
#define HAVE_TDM 1
#else
#define HAVE_TDM 0
#endif

// ---------------------------------------------------------------------------
// Bucketed exp: exact replay of the reference's greedy threshold decomposition.
// Greedy "subtract largest threshold <= v, multiply factor" == a single
// descending sweep with each level fired at most twice (all thr ratios < 3).
// Same mul/sub sequence as the reference loop -> bitwise-identical result.
// ---------------------------------------------------------------------------
__device__ __forceinline__ float bucket_eval(float x) {
  // POS: thresholds (desc) and their factors (fac[idx] for idx = level+1)
  const float PT[11] = {5.542f, 2.7726f, 1.3863f, 0.6931f, 0.4055f, 0.2231f,
                        0.1178f, 0.0606f, 0.0308f, 0.0155f, 0.0078f};
  const float PF[11] = {256.0f, 16.0f, 4.0f, 2.0f, 1.5f, 1.25f,
                        1.125f, 1.0625f, 1.03125f, 1.015625f, 1.0078125f};
  // NEG path (x < 0): thresholds on |x|, factors approximate exp(-thr)
  const float NT[11] = {5.542f, 2.7726f, 1.3863f, 0.6931f, 0.2877f, 0.1335f,
                        0.0645f, 0.0317f, 0.0157f, 0.0078f, 0.0039f};
  const float NF[11] = {0.00390625f, 0.0625f, 0.25f, 0.5f, 0.75f, 0.875f,
                        0.9375f, 0.96875f, 0.984375f, 0.9921875f, 0.99609375f};

  const bool neg = !(x >= 0.0f);     // matches reference split (x>=0 -> pos path)
  float v = fabsf(x);                // pos: v=x ; neg: v=|x|
  float o = 1.0f;

#pragma unroll
  for (int k = 0; k < 11; ++k) {
    const float t = neg ? NT[k] : PT[k];   // top 4 levels fold (PT==NT there)
    const float f = neg ? NF[k] : PF[k];
    // fire level up to 2x (branchless -> v_cmp + v_cndmask)
    {
      const bool  c  = v >= t;
      const float vm = v - t;
      const float om = o * f;
      v = c ? vm : v;
      o = c ? om : o;
    }
    {
      const bool  c  = v >= t;
      const float vm = v - t;
      const float om = o * f;
      v = c ? vm : v;
      o = c ? om : o;
    }
  }
  return o;
}

__device__ __forceinline__ f32x4 bucket_eval4(f32x4 a) {
  f32x4 r;
  r.x = bucket_eval(a.x);
  r.y = bucket_eval(a.y);
  r.z = bucket_eval(a.z);
  r.w = bucket_eval(a.w);
  return r;
}

#if HAVE_TDM
// ---------------------------------------------------------------------------
// Issue one TDM tensor_load_to_lds of a 256x16 fp32 tile (contiguous 16KB).
// D# layout per CDNA5 ISA 8.3/8.4: group0 {count=1, lds_addr, global_addr,
// type=2}; group1 {data_size=4B, tensor_dim0=256, tensor_dim1=16,
// tile_dim0=256, tile_dim1=16, dim0_stride=256}.
// ---------------------------------------------------------------------------
__device__ __forceinline__ void tdm_issue_tile(u64 gaddr, u32 lds_byte_addr) {
  u32x4 g0;
  g0.x = 1u;                                        // count=1, user mode
  g0.y = lds_byte_addr;                             // lds_addr [63:32]
  g0.z = (u32)(gaddr & 0xffffffffull);              // global_addr [95:64]
  g0.w = (u32)((gaddr >> 32) & 0x1ffffffull)        // global_addr [120:96]
       | (2u << 30);                                // type=2 ("image")
  i32x8 g1;
  g1.s0 = (int)(2u << 16);                          // data_size = 4 bytes
  g1.s1 = (int)((u32)TILE_COLS << 16);              // tensor_dim0[15:0]
  g1.s2 = (int)((u32)TILE_ROWS << 16);              // tensor_dim1[15:0]
  g1.s3 = (int)((u32)TILE_COLS << 16);              // tile_dim0
  g1.s4 = (int)TILE_ROWS;                           // tile_dim1 (tile_dim2=0)
  g1.s5 = (int)TILE_COLS;                           // tensor_dim0_stride[31:0]
  g1.s6 = 0;                                        // stride hi / dim1_stride lo
  g1.s7 = 0;
  i32x4 gz = {0, 0, 0, 0};                          // groups 2/3: 2D tensor
#if defined(__clang_major__) && (__clang_major__ >= 23)
  i32x8 gz8 = {0, 0, 0, 0, 0, 0, 0, 0};
  __builtin_amdgcn_tensor_load_to_lds(g0, g1, gz, gz, gz8, 0);
#else
  __builtin_amdgcn_tensor_load_to_lds(g0, g1, gz, gz, 0);
#endif
}
#endif // HAVE_TDM

// ---------------------------------------------------------------------------
// Main kernel: grid-stride over 4096-element tiles.
// TDM path: wave 0 DMAs tile t+stride into the spare LDS buffer while all
// waves compute on tile t (double buffered, s_wait_tensorcnt + barriers).
// All intra-block traffic is B128 (ds_load_b128 in, NT global_store_b128 out).
// ---------------------------------------------------------------------------
__global__ void __launch_bounds__(BLOCK)
exp_bucket_tdm_kernel(const float* __restrict__ x, float* __restrict__ y,
                      int ntiles) {
#if HAVE_TDM
  __shared__ __align__(16) float smem[2][TILE_ELEMS];
  const int tid = threadIdx.x;
  // scalar (SGPR) wave id -> guaranteed uniform branch around the TDM issue
  const int wid = __builtin_amdgcn_readfirstlane((int)(threadIdx.x >> 5));
  const int nb  = gridDim.x;
  const int t0  = blockIdx.x;
  if (t0 >= ntiles) return;

  const u64 xbase = (u64)(uintptr_t)x;

  if (wid == 0) {
    tdm_issue_tile(xbase + (u64)t0 * (TILE_ELEMS * 4ull),
                   (u32)(uintptr_t)&smem[0][0]);
  }

  int parity = 0;
  for (int t = t0; t < ntiles; t += nb) {
    const int tn = t + nb;
    if (wid == 0) {
      if (tn < ntiles) {
        tdm_issue_tile(xbase + (u64)tn * (TILE_ELEMS * 4ull),
                       (u32)(uintptr_t)&smem[parity ^ 1][0]);
        __builtin_amdgcn_s_wait_tensorcnt(1);  // oldest (tile t) resident
      } else {
        __builtin_amdgcn_s_wait_tensorcnt(0);
      }
    }
    __syncthreads();                            // tile t visible to all waves

    const f32x4* sv = (const f32x4*)&smem[parity][0];
    f32x4*       yv = (f32x4*)(y + (u64)t * TILE_ELEMS);
#pragma unroll 1
    for (int j = 0; j < TILE_VEC4 / BLOCK; ++j) {   // 4 iterations
      const int   idx = j * BLOCK + tid;            // coalesced b128 lanes
      const f32x4 a   = sv[idx];
      const f32x4 r   = bucket_eval4(a);            // 4 independent chains
      __builtin_nontemporal_store(r, &yv[idx]);     // TH=NT: don't thrash L2
    }
    __syncthreads();                            // buf free before next overwrite
    parity ^= 1;
  }
#else
  // Fallback: direct global streaming (no TDM builtin on this toolchain)
  const int tid = threadIdx.x;
  for (int t = blockIdx.x; t < ntiles; t += gridDim.x) {
    const f32x4* xv = (const f32x4*)(x + (u64)t * TILE_ELEMS);
    f32x4*       yv = (f32x4*)(y + (u64)t * TILE_ELEMS);
#pragma unroll 1
    for (int j = 0; j < TILE_VEC4 / BLOCK; ++j) {
      const int idx = j * BLOCK + tid;
      const f32x4 r = bucket_eval4(xv[idx]);
      __builtin_nontemporal_store(r, &yv[idx]);
    }
  }
#endif
}

// Tail elements (N % 4096) — N = 26,112,000 is an exact multiple, but stay
// generic so the kernel is correct for any size.
__global__ void __launch_bounds__(BLOCK)
exp_bucket_tail_kernel(const float* __restrict__ x, float* __restrict__ y,
                       int start, int n) {
  const int i = start + (int)(blockIdx.x * blockDim.x + threadIdx.x);
  if (i < n) y[i] = bucket_eval(x[i]);
}

extern "C" void kernel_launch(void* const* d_in, const int* in_sizes, int n_in,
                              void* d_out, int out_size, void* d_ws, size_t ws_size,
                              hipStream_t stream) {
  (void)n_in; (void)out_size; (void)d_ws; (void)ws_size;
  const float* x = (const float*)d_in[0];
  float*       y = (float*)d_out;
  const int n = in_sizes[0];

  const int ntiles = n / TILE_ELEMS;
  if (ntiles > 0) {
    int blocks = ntiles < 1024 ? ntiles : 1024;
    exp_bucket_tdm_kernel<<<dim3(blocks), dim3(BLOCK), 0, stream>>>(x, y, ntiles);
  }
  const int start = ntiles * TILE_ELEMS;
  const int tail  = n - start;
  if (tail > 0) {
    int blocks = (tail + BLOCK - 1) / BLOCK;
    exp_bucket_tail_kernel<<<dim3(blocks), dim3(BLOCK), 0, stream>>>(x, y, start, n);
  }
}